// FaceAttention_50525995270597
// MI455X (gfx1250) — compile-verified
//
#include <hip/hip_runtime.h>

#define MAXF 15
#define DDIM 512
#define ROWS 16          // rows (samples) per block
#define WAVES 8
#define THREADS 256
#define SSTRIDE 516      // LDS row stride in floats: 516 % 64 == 4 -> conflict-free column reads

typedef float v2f __attribute__((ext_vector_type(2)));
typedef float v4f __attribute__((ext_vector_type(4)));
typedef float v8f __attribute__((ext_vector_type(8)));

__global__ __launch_bounds__(THREADS, 2)
void face_avg_wmma_kernel(const float* __restrict__ X,      // [B, 15, 512]
                          const float* __restrict__ W,      // [512, 3]
                          const float* __restrict__ bias,   // [3]
                          const int*   __restrict__ nFaces, // [B]
                          float* __restrict__ out,          // [B, 3]
                          int Btotal)
{
    __shared__ float sS[ROWS * SSTRIDE];   // masked-mean rows, 16 x 512 (padded stride)
    __shared__ float sW4[DDIM * 4];        // W padded to 4 cols (col 3 = 0), K-pair interleaved:
                                           //   element (k, c) at (k>>1)*8 + c*2 + (k&1)
    __shared__ float sPart[WAVES][ROWS][4];// per-wave partial C; col 3 is a write-dump, never read
    __shared__ int   sN[ROWS];

    const int tid  = threadIdx.x;
    const int lane = tid & 31;
    const int wave = tid >> 5;
    const int rowBase = blockIdx.x * ROWS;

    // Stage W into LDS, zero-padded 4th column, pair-interleaved for b64 B-fragment loads.
    for (int i = tid; i < DDIM * 4; i += THREADS) {
        const int d = i >> 2, c = i & 3;
        const float v = (c < 3) ? W[d * 3 + c] : 0.0f;
        sW4[(d >> 1) * 8 + c * 2 + (d & 1)] = v;
    }

    // ---- Phase 1: bandwidth-bound masked mean over faces (each wave: 2 rows) ----
    for (int rr = 0; rr < 2; ++rr) {
        const int r   = wave * 2 + rr;            // 0..15
        const int row = rowBase + r;
        int n = 0;
        if (row < Btotal) {
            n = nFaces[row];
            n = n > MAXF ? MAXF : n;
        }
        if (lane == 0) sN[r] = n;

        v4f a0 = {}, a1 = {}, a2 = {}, a3 = {};
        const float* rp = X + (size_t)row * (MAXF * DDIM);
        for (int f = 0; f < n; ++f) {             // n is wave-uniform
            const v4f* fp = (const v4f*)(rp + f * DDIM);
            a0 += __builtin_nontemporal_load(fp + lane);
            a1 += __builtin_nontemporal_load(fp + lane + 32);
            a2 += __builtin_nontemporal_load(fp + lane + 64);
            a3 += __builtin_nontemporal_load(fp + lane + 96);
        }
        const float inv = (n > 0) ? (1.0f / (float)n) : 0.0f;
        a0 *= inv; a1 *= inv; a2 *= inv; a3 *= inv;

        float* sp = &sS[r * SSTRIDE];
        *(v4f*)(sp + 4 * lane)       = a0;        // d = 4*lane + 0..3
        *(v4f*)(sp + 4 * lane + 128) = a1;
        *(v4f*)(sp + 4 * lane + 256) = a2;
        *(v4f*)(sp + 4 * lane + 384) = a3;
    }
    __syncthreads();

    // ---- Phase 2: S[16x512] @ W[512x3] via v_wmma_f32_16x16x4_f32 ----
    // Wave w covers K = w*64 .. w*64+63 (16 k-steps of 4), own accumulator.
    // A 16x4 f32 layout: lanes 0-15 -> M=lane, K = kk+{0,1}; lanes 16-31 -> M=lane-16, K = kk+{2,3}
    // B 4x16  f32 layout: lanes 0-15 -> N=lane, K = kk+{0,1}; lanes 16-31 -> N=lane-16, K = kk+{2,3}
    v8f c = {};
    const int m16  = lane & 15;                   // M for A, N for B
    const int bcol = m16 < 3 ? m16 : 3;           // N>=3 lanes read the zero column (broadcast)
    const int koff = (lane >> 4) << 1;            // 0 or 2
    const int kw   = wave * 64;
    #pragma unroll
    for (int j = 0; j < 16; ++j) {
        const int kk = kw + j * 4;
        const int k0 = kk + koff;                 // even
        v2f a = *(const v2f*)&sS[m16 * SSTRIDE + k0];                 // ds_load_b64
        v2f bm = *(const v2f*)&sW4[(k0 >> 1) * 8 + bcol * 2];         // ds_load_b64 (k0, k0+1)
        c = __builtin_amdgcn_wmma_f32_16x16x4_f32(
                /*neg_a=*/false, a, /*neg_b=*/false, bm,
                /*c_mod=*/(short)0, c, /*reuse_a=*/false, /*reuse_b=*/false);
    }

    // C layout: VGPR i -> lanes 0-15 hold (M=i, N=lane), lanes 16-31 hold (M=i+8, N=lane-16).
    // Lanes with N>=3 dump into padding col 3 (unordered same-address writes, never read).
    const int mhi = (lane >> 4) << 3;             // 0 or 8
    #pragma unroll
    for (int i = 0; i < 8; ++i) {
        sPart[wave][i + mhi][bcol] = c[i];
    }
    __syncthreads();

    // ---- Phase 3: reduce 8 wave partials, apply bias / zero-face epilogue ----
    if (tid < ROWS * 3) {
        const int M = tid / 3, cc = tid % 3;
        float acc = 0.0f;
        #pragma unroll
        for (int w = 0; w < WAVES; ++w) acc += sPart[w][M][cc];
        const int row = rowBase + M;
        if (row < Btotal) {
            const int n = sN[M];
            out[(size_t)row * 3 + cc] = (n > 0) ? (acc + bias[cc]) : 0.0f;
        }
    }
}

extern "C" void kernel_launch(void* const* d_in, const int* in_sizes, int n_in,
                              void* d_out, int out_size, void* d_ws, size_t ws_size,
                              hipStream_t stream) {
    const float* X    = (const float*)d_in[0];   // [B,15,512]
    const float* W    = (const float*)d_in[1];   // [512,3]
    const float* bias = (const float*)d_in[2];   // [3]
    const int*   nF   = (const int*)d_in[3];     // [B]
    // d_in[4] = labels, unused by the forward output
    float* out = (float*)d_out;
    const int B = in_sizes[3];
    const int grid = (B + ROWS - 1) / ROWS;
    face_avg_wmma_kernel<<<grid, THREADS, 0, stream>>>(X, W, bias, nF, out, B);
}